// DirectVG_33535104647248
// MI455X (gfx1250) — compile-verified
//
#include <hip/hip_runtime.h>
#include <stdint.h>

// Problem constants (match reference): B=64, N=1000, M=100, ITERS=5 -> T=6 states.
#define BB 64
#define NN 1000
#define MM 100
#define TT 6
#define BLK 256
#define LR_POS 0.45f
#define LR_SIZE 0.4f

// One workgroup handles (batch b, a chunk of up to 256 boxes).
// Phase 0: async-load the 100 gt boxes (1.6KB) into LDS (global_load_async_to_lds_b128),
//          then precompute the 100 gt areas into LDS.
// Per round t: each thread computes its 100-float IoU row into an LDS tile whose
// layout is a byte-exact image of the global [rows,100] tile, tracks argmax, writes
// its box as a coalesced float4; then the whole block bulk-copies the LDS tile to
// global with global_store_async_from_lds_b128 (16B/lane, fully coalesced),
// synchronized via s_wait_asynccnt + barriers so LDS can be reused next round.
__global__ __launch_bounds__(BLK) void box_refine_kernel(
    const float* __restrict__ boxes,     // [B,N,4]
    const float* __restrict__ gt_boxes,  // [B,M,4]
    float* __restrict__ out_boxes,       // [B,T,N,4]
    float* __restrict__ out_sims)        // [B,T,N,M]
{
    __shared__ __align__(16) float s_gt[MM * 4];        // 1.6 KB gt boxes
    __shared__ __align__(16) float s_ga[MM];            // 400 B precomputed gt areas
    __shared__ __align__(16) float s_sims[BLK * MM];    // 100 KB, exact image of global tile

    const int b   = blockIdx.y;
    const int n0  = blockIdx.x * BLK;
    const int tid = threadIdx.x;
    const int rows = min(BLK, NN - n0);

    // ---- async broadcast of gt tile into LDS ----
    if (tid < MM) {
        unsigned loff = (unsigned)(uintptr_t)(&s_gt[tid * 4]);          // low 32 bits == LDS byte offset
        uint64_t gaddr = (uint64_t)(uintptr_t)(gt_boxes + ((size_t)b * MM + tid) * 4);
        asm volatile("global_load_async_to_lds_b128 %0, %1, off"
                     :: "v"(loff), "v"(gaddr) : "memory");
    }
    asm volatile("s_wait_asynccnt 0x0" ::: "memory");
    __syncthreads();

    // ---- precompute gt areas (loop-invariant across all 6 rounds) ----
    if (tid < MM) {
        const float gx1 = s_gt[tid * 4 + 0], gy1 = s_gt[tid * 4 + 1];
        const float gx2 = s_gt[tid * 4 + 2], gy2 = s_gt[tid * 4 + 3];
        s_ga[tid] = (gx2 - gx1) * (gy2 - gy1);
    }
    __syncthreads();

    const int  n      = n0 + tid;
    const bool active = (tid < rows);

    float x1 = 0.f, y1 = 0.f, x2 = 1.f, y2 = 1.f;
    if (active) {
        const float4 bb = *(const float4*)(boxes + ((size_t)b * NN + n) * 4);
        x1 = bb.x; y1 = bb.y; x2 = bb.z; y2 = bb.w;
    }
    int best = 0;

    for (int t = 0; t < TT; ++t) {
        if (active) {
            if (t > 0) {
                // move box toward best-IoU gt from previous round
                const float gx1 = s_gt[best * 4 + 0], gy1 = s_gt[best * 4 + 1];
                const float gx2 = s_gt[best * 4 + 2], gy2 = s_gt[best * 4 + 3];
                const float cx  = 0.5f * (x1 + x2),  cy  = 0.5f * (y1 + y2);
                const float gcx = 0.5f * (gx1 + gx2), gcy = 0.5f * (gy1 + gy2);
                const float dcx = gcx - cx, dcy = gcy - cy;
                const float dsx = (gx2 - gx1) - (x2 - x1);
                const float dsy = (gy2 - gy1) - (y2 - y1);
                const float nx1 = x1 + LR_POS * dcx;
                const float ny1 = y1 + LR_POS * dcy;
                const float nx2 = x2 + LR_POS * dcx + LR_SIZE * (dsx - dcx);
                const float ny2 = y2 + LR_POS * dcy + LR_SIZE * (dsy - dcy);
                x1 = nx1; y1 = ny1; x2 = nx2; y2 = ny2;
            }
            const float area = (x2 - x1) * (y2 - y1);
            float bestv = -__builtin_huge_valf();
            int   bi    = 0;
            float* srow = &s_sims[tid * MM];
            #pragma unroll 4
            for (int m = 0; m < MM; ++m) {
                const float gx1 = s_gt[m * 4 + 0], gy1 = s_gt[m * 4 + 1];
                const float gx2 = s_gt[m * 4 + 2], gy2 = s_gt[m * 4 + 3];
                const float lx  = fmaxf(x1, gx1), ly = fmaxf(y1, gy1);
                const float rx  = fminf(x2, gx2), ry = fminf(y2, gy2);
                const float w   = fmaxf(rx - lx, 0.f);
                const float h   = fmaxf(ry - ly, 0.f);
                const float inter = w * h;
                const float den   = (area + s_ga[m]) - inter;   // strictly > 0
                // v_rcp_f32 + one Newton-Raphson step (~1-2 ulp) instead of IEEE div chain
                float r = __builtin_amdgcn_rcpf(den);
                r = r * fmaf(-den, r, 2.0f);
                const float iou = inter * r;
                srow[m] = iou;
                if (iou > bestv) { bestv = iou; bi = m; }   // first-max == jnp.argmax
            }
            best = bi;
            // coalesced float4-per-lane box output
            float4 ob; ob.x = x1; ob.y = y1; ob.z = x2; ob.w = y2;
            *(float4*)(out_boxes + (((size_t)b * TT + t) * NN + n) * 4) = ob;
        }
        __syncthreads();   // sims tile fully written in LDS

        // ---- bulk async LDS -> global copy of the sims tile (16B per lane) ----
        {
            float* gbase = out_sims + (((size_t)b * TT + t) * NN + n0) * MM;
            const int chunks = rows * (MM / 4);   // 100 floats/row = 25 b128 chunks
            for (int c = tid; c < chunks; c += BLK) {
                unsigned loff  = (unsigned)(uintptr_t)(&s_sims[c * 4]);
                uint64_t gaddr = (uint64_t)(uintptr_t)(gbase + (size_t)c * 4);
                asm volatile("global_store_async_from_lds_b128 %0, %1, off"
                             :: "v"(gaddr), "v"(loff) : "memory");
            }
            asm volatile("s_wait_asynccnt 0x0" ::: "memory");
        }
        __syncthreads();   // LDS tile free for next round
    }
}

extern "C" void kernel_launch(void* const* d_in, const int* in_sizes, int n_in,
                              void* d_out, int out_size, void* d_ws, size_t ws_size,
                              hipStream_t stream) {
    const float* boxes    = (const float*)d_in[0];  // [64,1000,4]
    const float* gt_boxes = (const float*)d_in[1];  // [64,100,4]
    float* out       = (float*)d_out;
    float* out_boxes = out;                                   // [64,6,1000,4]
    float* out_sims  = out + (size_t)BB * TT * NN * 4;        // [64,6,1000,100]

    dim3 grid((NN + BLK - 1) / BLK, BB);   // 4 x 64 workgroups
    box_refine_kernel<<<grid, BLK, 0, stream>>>(boxes, gt_boxes, out_boxes, out_sims);
}